// SimpleRNN_19971597926678
// MI455X (gfx1250) — compile-verified
//
#include <hip/hip_runtime.h>
#include <math.h>

typedef __attribute__((ext_vector_type(2))) float v2f;
typedef __attribute__((ext_vector_type(4))) float v4f;
typedef __attribute__((ext_vector_type(8))) float v8f;

#define HDIM 128
#define TM 16
#define LDS_STRIDE 132   // padded row stride in floats (8B/16B aligned, avoids bank conflicts)

// D = A(16x4) * B(4x16) + C, full fp32 WMMA
__device__ __forceinline__ v8f wmma4(v2f a, v2f b, v8f c) {
    return __builtin_amdgcn_wmma_f32_16x16x4_f32(false, a, false, b, (short)0, c, false, false);
}

__device__ __forceinline__ float act_relu_tanh(float z) {
#if __has_builtin(__builtin_amdgcn_tanhf)
    float v = __builtin_amdgcn_tanhf(z);   // hardware V_TANH_F32
#else
    float v = tanhf(z);
#endif
    return fmaxf(v, 0.f);                  // relu(tanh(.))
}

// Preload this wave's 128x16 slice of W (row-major [128][128]) into 32 v2f fragments.
// B-matrix 4x16 layout per k-step: lanes 0-15 hold rows K=4k,4k+1 (v0,v1) at N=lane;
// lanes 16-31 hold rows K=4k+2,4k+3 at N=lane-16.
__device__ __forceinline__ void load_wfrags(const float* __restrict__ W,
                                            int n0, int col16, int hi, v2f* bfrag) {
    const float* wp = W + (hi * 2) * HDIM + n0 + col16;
#pragma unroll
    for (int k = 0; k < 32; ++k) {
        bfrag[k].x = wp[(4 * k) * HDIM];
        bfrag[k].y = wp[(4 * k + 1) * HDIM];
    }
}

// 16x16 tile of X(16x128) @ W(128x128): load all A frags as one LDS clause,
// then 4 independent WMMA chains of depth 8.
__device__ __forceinline__ v8f tile_matmul(const float* __restrict__ arow,
                                           const v2f* __restrict__ bfrag) {
    v2f a[32];
#pragma unroll
    for (int k = 0; k < 32; ++k)
        a[k] = *(const v2f*)(arow + 4 * k);
    v8f c0 = {}, c1 = {}, c2 = {}, c3 = {};
#pragma unroll
    for (int k = 0; k < 8; ++k) {
        c0 = wmma4(a[k],      bfrag[k],      c0);
        c1 = wmma4(a[k + 8],  bfrag[k + 8],  c1);
        c2 = wmma4(a[k + 16], bfrag[k + 16], c2);
        c3 = wmma4(a[k + 24], bfrag[k + 24], c3);
    }
    return (c0 + c1) + (c2 + c3);
}

// ---------------- Phase 1 & 3: Y[M,128] = X[M,128] @ W[128,128] + bias ----------------
__global__ __launch_bounds__(256) void gemm128_bias(const float* __restrict__ X,
                                                    const float* __restrict__ W,
                                                    const float* __restrict__ bias,
                                                    float* __restrict__ Y,
                                                    int M) {
    __shared__ float xs[TM * LDS_STRIDE];
    const int tid   = threadIdx.x;
    const int lane  = tid & 31;
    const int wv    = tid >> 5;      // 0..7 -> N tile
    const int n0    = wv * 16;
    const int col16 = lane & 15;
    const int hi    = lane >> 4;     // 0 or 1

    v2f bfrag[32];
    load_wfrags(W, n0, col16, hi, bfrag);
    const float bval = bias[n0 + col16];

    // A-matrix 16x4 layout: lanes 0-15 row M=lane, K=4k..4k+1; lanes 16-31 K=4k+2..4k+3
    const float* arow = &xs[col16 * LDS_STRIDE + hi * 2];

    const int nstrips = M / TM;
    for (int strip = blockIdx.x; strip < nstrips; strip += gridDim.x) {
        const float* xp = X + (size_t)strip * (TM * HDIM);
        // Cooperative load of contiguous 16x128 strip (8KB) into padded LDS
#pragma unroll
        for (int i = 0; i < 2; ++i) {
            int idx = tid + i * 256;       // float4 index 0..511
            int lin = idx * 4;
            int r = lin >> 7;
            int c = lin & 127;
            *(v4f*)(&xs[r * LDS_STRIDE + c]) = *(const v4f*)(xp + lin);
        }
        __syncthreads();

        v8f acc = tile_matmul(arow, bfrag);

        // C/D layout: VGPR i -> (M = i + 8*hi, N = n0 + col16)
        float* yp = Y + (size_t)strip * (TM * HDIM) + (size_t)(8 * hi) * HDIM + n0 + col16;
#pragma unroll
        for (int i = 0; i < 8; ++i)
            yp[(size_t)i * HDIM] = acc[i] + bval;
        __syncthreads();
    }
}

// ---------------- Phase 2: serial recurrence, h_t = relu(tanh(xi_t + h@Wh)) ----------------
// One WG per 16 batch rows; h lives in LDS; Wh slice lives in 64 VGPRs per lane.
// ws holds xi[B,S,H] on entry; h_t overwrites xi_t in place.
__global__ __launch_bounds__(256) void rnn_recur(const float* __restrict__ Wh,
                                                 float* __restrict__ ws,
                                                 int S) {
    __shared__ float hs[TM * LDS_STRIDE];
    const int tid   = threadIdx.x;
    const int lane  = tid & 31;
    const int wv    = tid >> 5;
    const int n0    = wv * 16;
    const int col16 = lane & 15;
    const int hi    = lane >> 4;
    const int b0    = blockIdx.x * TM;

    v2f bfrag[32];
    load_wfrags(Wh, n0, col16, hi, bfrag);

    for (int i = tid; i < TM * LDS_STRIDE; i += 256) hs[i] = 0.f;
    __syncthreads();

    const size_t rowstride = (size_t)S * HDIM;             // stride between batch rows
    // This lane owns elements (b = b0 + i + 8*hi, n = n0 + col16) for i = 0..7
    float* gp = ws + (size_t)(b0 + 8 * hi) * rowstride + n0 + col16;
    const float* arow = &hs[col16 * LDS_STRIDE + hi * 2];

    // Software pipeline: xi for step t prefetched one iteration ahead
    float xi_cur[8];
#pragma unroll
    for (int i = 0; i < 8; ++i) xi_cur[i] = gp[(size_t)i * rowstride];

    for (int t = 0; t < S; ++t) {
        float xi_nxt[8];
        if (t + 1 < S) {
#pragma unroll
            for (int i = 0; i < 8; ++i)
                xi_nxt[i] = gp[(size_t)i * rowstride + (size_t)(t + 1) * HDIM];
        } else {
#pragma unroll
            for (int i = 0; i < 8; ++i) xi_nxt[i] = 0.f;
        }

        v8f acc = tile_matmul(arow, bfrag);
        __syncthreads();   // all waves done reading old h from LDS

#pragma unroll
        for (int i = 0; i < 8; ++i) {
            float v = act_relu_tanh(acc[i] + xi_cur[i]);
            hs[(i + 8 * hi) * LDS_STRIDE + n0 + col16] = v;      // new h for next step
            gp[(size_t)i * rowstride + (size_t)t * HDIM] = v;    // stream h_t out (over xi_t)
        }
        __syncthreads();   // new h visible to all waves

#pragma unroll
        for (int i = 0; i < 8; ++i) xi_cur[i] = xi_nxt[i];
    }
}

extern "C" void kernel_launch(void* const* d_in, const int* in_sizes, int n_in,
                              void* d_out, int out_size, void* d_ws, size_t ws_size,
                              hipStream_t stream) {
    const float* x  = (const float*)d_in[0];   // [B,S,I]
    const float* Wi = (const float*)d_in[1];   // [I,H]
    const float* Wh = (const float*)d_in[2];   // [H,H]
    const float* b  = (const float*)d_in[3];   // [H]
    const float* Wo = (const float*)d_in[4];   // [H,O]
    const float* bo = (const float*)d_in[5];   // [O]
    float* y  = (float*)d_out;                 // [B,S,O]
    float* ws = (float*)d_ws;                  // [B,S,H] fp32 = 64MB scratch

    const int B = 64, S = 2048;
    const int M = B * S;                       // 131072 rows

    // Phase 1: xi = x @ Wi + b  -> ws
    gemm128_bias<<<1024, 256, 0, stream>>>(x, Wi, b, ws, M);
    // Phase 2: serial recurrence, h_t overwrites xi_t in ws
    rnn_recur<<<B / TM, 256, 0, stream>>>(Wh, ws, S);
    // Phase 3: y = hs @ Wo + bo
    gemm128_bias<<<1024, 256, 0, stream>>>(ws, Wo, bo, y, M);
}